// PointNet_Plus_18056042513006
// MI455X (gfx1250) — compile-verified
//
#include <hip/hip_runtime.h>
#include <cstdint>

// ---------------------------------------------------------------------------
// PointNet++ pipeline for MI455X (gfx1250, wave32, WMMA).
// All GEMMs use V_WMMA_F32_16X16X4_F32 (fp32 matrix pipe) to match the fp32
// reference exactly; MLP stages are fused with LDS-resident activations.
// Activations live in LDS in a B-operand-packed layout so each WMMA B operand
// is a single aligned ds_load_b64 (no register shuffling):
//   Xp[(k>>2)*S + (col<<2) + (k&3)],  S = 4*P + 8  (bank-conflict-free pad)
// Each wave register-blocks a 16x64 output strip (4 accumulators) so one
// packed-A global load feeds 4 WMMAs.
// ---------------------------------------------------------------------------

typedef __attribute__((ext_vector_type(2))) float v2f;
typedef __attribute__((ext_vector_type(8))) float v8f;

__device__ __forceinline__ v8f wmma_f32(v2f a, v2f b, v8f c) {
  // D = A(16x4) * B(4x16) + C(16x16), fp32.
  return __builtin_amdgcn_wmma_f32_16x16x4_f32(
      /*neg_a=*/false, a, /*neg_b=*/false, b,
      /*c_mod=*/(short)0, c, /*reuse_a=*/false, /*reuse_b=*/false);
}

// One GEMM layer: Y(Cout x P) = relu(Wp @ X + bias). X and Y are LDS buffers
// in the B-packed layout (slice stride S dwords per 4 channels). Wp is
// pre-packed in WMMA A-operand order:
//   Wp[(m_tile*(CinP/4) + kstep)*64 + lane*2 + i]
//     <- W[m_tile*16 + (lane&15)][kstep*4 + (lane>>4)*2 + i]
__device__ void mlp_gemm_lds(float* __restrict__ Y, const float* __restrict__ Xl,
                             const float* __restrict__ Wp, const float* __restrict__ bias,
                             int Cout, int CinP, int P, int S) {
  const int lane = threadIdx.x & 31;
  const int wave = threadIdx.x >> 5;
  const int nw = blockDim.x >> 5;
  const int half = lane >> 4;
  const int l16 = lane & 15;
  const int ksteps = CinP >> 2;
  const int groupsN = P >> 6;                 // strips of 64 columns (P is 64 or 128)
  const int ngroups = (Cout >> 4) * groupsN;
  for (int g = wave; g < ngroups; g += nw) {
    const int mt = g / groupsN;
    const int n0 = (g - mt * groupsN) << 6;
    const float* wp = Wp + ((size_t)mt * ksteps) * 64 + lane * 2;
    const float* xp = Xl + ((n0 + l16) << 2) + half * 2;
    v8f acc0 = {}, acc1 = {}, acc2 = {}, acc3 = {};
    for (int kk = 0; kk < ksteps; ++kk) {
      v2f a = *(const v2f*)wp;                // one coalesced A load feeds 4 WMMAs
      v2f b0 = *(const v2f*)(xp);             // single ds_load_b64 per B operand
      v2f b1 = *(const v2f*)(xp + 64);
      v2f b2 = *(const v2f*)(xp + 128);
      v2f b3 = *(const v2f*)(xp + 192);
      acc0 = wmma_f32(a, b0, acc0);
      acc1 = wmma_f32(a, b1, acc1);
      acc2 = wmma_f32(a, b2, acc2);
      acc3 = wmma_f32(a, b3, acc3);
      wp += 64;
      xp += S;
    }
    const int rowbase = (mt << 4) + half * 8;
    const int colq = (n0 + l16) << 2;
#pragma unroll
    for (int v = 0; v < 8; v += 2) {
      const int row = rowbase + v;
      const float bv0 = bias[row];
      const float bv1 = bias[row + 1];
      float* yp = Y + (row >> 2) * S + colq + (row & 3);
      v2f p;
      p.x = fmaxf(acc0[v] + bv0, 0.0f);
      p.y = fmaxf(acc0[v + 1] + bv1, 0.0f);
      *(v2f*)(yp) = p;
      p.x = fmaxf(acc1[v] + bv0, 0.0f);
      p.y = fmaxf(acc1[v + 1] + bv1, 0.0f);
      *(v2f*)(yp + 64) = p;
      p.x = fmaxf(acc2[v] + bv0, 0.0f);
      p.y = fmaxf(acc2[v + 1] + bv1, 0.0f);
      *(v2f*)(yp + 128) = p;
      p.x = fmaxf(acc3[v] + bv0, 0.0f);
      p.y = fmaxf(acc3[v + 1] + bv1, 0.0f);
      *(v2f*)(yp + 192) = p;
    }
  }
}

// ---------------------------------------------------------------------------
// Fused stage-1: 3-layer pointwise MLP over (Bx, Cin0, Mtot, 64) + max over k.
// One workgroup handles PM=2 m-values (P=128 positions) fully in LDS.
// Output pooled rows written at Out[b][outRowOff + c][m].
// ---------------------------------------------------------------------------
__global__ __launch_bounds__(256) void mlp3_pool_k(
    const float* __restrict__ X, float* __restrict__ Out,
    const float* __restrict__ W1, const float* __restrict__ b1, int C1,
    const float* __restrict__ W2, const float* __restrict__ b2, int C2,
    const float* __restrict__ W3, const float* __restrict__ b3, int C3,
    int Cin0, int Mtot, int outLD, int outBatchStride, int outRowOff) {
  extern __shared__ float lds[];
  const int P = 128, S = 4 * 128 + 8, PM = 2;   // S = 520 dwords per 4-channel slice
  const int tid = threadIdx.x;
  const int b = blockIdx.y;
  const int mt = blockIdx.x;
  float* X0 = lds;
  float* Y1 = X0 + (size_t)(Cin0 >> 2) * S;
  float* Y2 = Y1 + (size_t)(C1 >> 2) * S;
  float* Y3 = (Cin0 + C1 >= C3) ? lds : (Y2 + (size_t)(C2 >> 2) * S);

  const size_t plane = (size_t)Mtot * 64;
  const size_t p0 = (size_t)mt * P;
  for (int t = tid; t < Cin0 * P; t += blockDim.x) {
    int c = t / P, col = t - c * P;
    X0[(c >> 2) * S + (col << 2) + (c & 3)] =
        X[((size_t)b * Cin0 + c) * plane + p0 + col];
  }
  __syncthreads();
  mlp_gemm_lds(Y1, X0, W1, b1, C1, Cin0, P, S);
  __syncthreads();
  mlp_gemm_lds(Y2, Y1, W2, b2, C2, C1, P, S);
  __syncthreads();
  mlp_gemm_lds(Y3, Y2, W3, b3, C3, C2, P, S);
  __syncthreads();
  for (int t = tid; t < C3 * PM; t += blockDim.x) {
    int c = t / PM, mi = t - c * PM;
    const float* p = Y3 + (c >> 2) * S + ((mi * 64) << 2) + (c & 3);
    float mx = p[0];
    for (int k2 = 1; k2 < 64; ++k2) mx = fmaxf(mx, p[k2 << 2]);
    Out[(size_t)b * outBatchStride + (size_t)(outRowOff + c) * outLD + (mt * PM + mi)] = mx;
  }
}

// ---------------------------------------------------------------------------
// Fused ball-query group + 3-layer MLP + max over the 64 group slots.
// One workgroup per (batch, center m). Pt: (Bx, 132, 512) with rows 0-2 = xyz,
// 3..130 = features, 131 = zero pad. Out: buf (Bx, 260, 128):
// rows 0-2 = center xyz, 3..258 = pooled features, 259 = zero pad.
// ---------------------------------------------------------------------------
__global__ __launch_bounds__(256) void group_mlp3_k(
    const float* __restrict__ Pt, float* __restrict__ Out,
    const float* __restrict__ W1, const float* __restrict__ b1, int C1,
    const float* __restrict__ W2, const float* __restrict__ b2, int C2,
    const float* __restrict__ W3, const float* __restrict__ b3, int C3) {
  extern __shared__ float lds[];
  const int S = 4 * 64 + 8;                  // 264 dwords per 4-channel slice
  const int CIN = 132, NPTS = 512, M2 = 128;
  const int tid = threadIdx.x;
  const int b = blockIdx.y;
  const int m = blockIdx.x;
  float* X0 = lds;                           // (132/4)*264 = 8712
  float* Y1 = X0 + (CIN >> 2) * S;           // (C1/4)*264
  float* Y2 = Y1 + (C1 >> 2) * S;            // (C2/4)*264
  float* d2s = Y2 + (C2 >> 2) * S;           // 512
  float* ctr = d2s + NPTS;                   // 4
  int* sel = (int*)(ctr + 4);                // 64
  float* Y3 = X0;                            // overlay: (C3/4)*S <= (CIN/4 + C1/4)*S

  const float* P0 = Pt + (size_t)b * CIN * NPTS;
  if (tid < 3) ctr[tid] = P0[(size_t)tid * NPTS + m];
  __syncthreads();
  const float cx = ctr[0], cy = ctr[1], cz = ctr[2];
  for (int n = tid; n < NPTS; n += blockDim.x) {
    float dx = P0[n] - cx;
    float dy = P0[NPTS + n] - cy;
    float dz = P0[2 * NPTS + n] - cz;
    d2s[n] = dx * dx + dy * dy + dz * dz;
  }
  __syncthreads();
  // Stable top-64 selection (matches jax.lax.top_k set semantics): rank by
  // (d2, index). Out-of-radius picks fall back to the center itself.
  for (int n = tid; n < NPTS; n += blockDim.x) {
    const float d = d2s[n];
    int rank = 0;
    for (int j = 0; j < NPTS; ++j) {
      float dj = d2s[j];
      rank += (dj < d || (dj == d && j < n)) ? 1 : 0;
    }
    if (rank < 64) sel[rank] = (d <= 0.25f) ? n : m;
  }
  __syncthreads();
  // Gather 132 x 64 tile (B-packed layout), centering the xyz channels.
  for (int t = tid; t < CIN * 64; t += blockDim.x) {
    int c = t >> 6, slot = t & 63;
    int idx = sel[slot];
    float v = P0[(size_t)c * NPTS + idx];
    if (c < 3) v -= ctr[c];
    X0[(c >> 2) * S + (slot << 2) + (c & 3)] = v;
  }
  __syncthreads();
  mlp_gemm_lds(Y1, X0, W1, b1, C1, CIN, 64, S);
  __syncthreads();
  mlp_gemm_lds(Y2, Y1, W2, b2, C2, C1, 64, S);
  __syncthreads();
  mlp_gemm_lds(Y3, Y2, W3, b3, C3, C2, 64, S);
  __syncthreads();
  float* out = Out + (size_t)b * 260 * M2;
  for (int c = tid; c < C3; c += blockDim.x) {
    const float* p = Y3 + (c >> 2) * S + (c & 3);
    float mx = p[0];
    for (int k2 = 1; k2 < 64; ++k2) mx = fmaxf(mx, p[k2 << 2]);
    out[(size_t)(3 + c) * M2 + m] = mx;
  }
  if (tid < 3) out[(size_t)tid * M2 + m] = ctr[tid];
  if (tid == 3) out[(size_t)259 * M2 + m] = 0.0f;
}

// ---------------------------------------------------------------------------
// Unfused batched GEMM (stage-3 + fc1): Y(Cout,N) = act(Wp @ X + b) per batch.
// grid = (ceil(N/256), Cout/16, Bx), block = 128 (4 waves, one 16x64 strip
// each, 4 accumulators sharing one A stream; per-subtile column guards).
// ---------------------------------------------------------------------------
__global__ __launch_bounds__(128) void gemm_glb_k(
    const float* __restrict__ Wp, const float* __restrict__ bias,
    const float* __restrict__ X, float* __restrict__ Y,
    int Cout, int CinP, int N, int doRelu) {
  const int lane = threadIdx.x & 31;
  const int wave = threadIdx.x >> 5;
  const int half = lane >> 4;
  const int l16 = lane & 15;
  const int mt = blockIdx.y;
  const int c0 = blockIdx.x * 256 + wave * 64 + l16;
  const bool ok0 = c0 < N;
  const bool ok1 = c0 + 16 < N;
  const bool ok2 = c0 + 32 < N;
  const bool ok3 = c0 + 48 < N;
  const float* Xb = X + (size_t)blockIdx.z * CinP * N;
  float* Yb = Y + (size_t)blockIdx.z * Cout * N;
  const int ksteps = CinP >> 2;
  const float* wp = Wp + (size_t)mt * ksteps * 64 + lane * 2;
  const float* xp = Xb + (size_t)(half * 2) * N + (ok0 ? c0 : 0);
  v8f acc0 = {}, acc1 = {}, acc2 = {}, acc3 = {};
  for (int kk = 0; kk < ksteps; ++kk) {
    v2f a = *(const v2f*)wp;
    v2f b0, b1, b2, b3;
    b0.x = ok0 ? xp[0] : 0.0f;       b0.y = ok0 ? xp[N] : 0.0f;
    b1.x = ok1 ? xp[16] : 0.0f;      b1.y = ok1 ? xp[N + 16] : 0.0f;
    b2.x = ok2 ? xp[32] : 0.0f;      b2.y = ok2 ? xp[N + 32] : 0.0f;
    b3.x = ok3 ? xp[48] : 0.0f;      b3.y = ok3 ? xp[N + 48] : 0.0f;
    acc0 = wmma_f32(a, b0, acc0);
    acc1 = wmma_f32(a, b1, acc1);
    acc2 = wmma_f32(a, b2, acc2);
    acc3 = wmma_f32(a, b3, acc3);
    wp += 64;
    xp += (size_t)4 * N;
  }
  const int row = (mt << 4) + half * 8;
#pragma unroll
  for (int v = 0; v < 8; ++v) {
    float bv = bias[row + v];
    float* yp = Yb + (size_t)(row + v) * N + c0;
    float v0 = acc0[v] + bv, v1 = acc1[v] + bv, v2 = acc2[v] + bv, v3 = acc3[v] + bv;
    if (doRelu) {
      v0 = fmaxf(v0, 0.0f); v1 = fmaxf(v1, 0.0f);
      v2 = fmaxf(v2, 0.0f); v3 = fmaxf(v3, 0.0f);
    }
    if (ok0) yp[0]  = v0;
    if (ok1) yp[16] = v1;
    if (ok2) yp[32] = v2;
    if (ok3) yp[48] = v3;
  }
}

// ---------------------------------------------------------------------------
// Utility kernels
// ---------------------------------------------------------------------------

// Pack W (Cout x Cin, row-major) into WMMA A-operand order with zero padding
// of the k dimension to CinP (multiple of 4).
__global__ void pack_w_k(const float* __restrict__ W, float* __restrict__ Wp,
                         int Cout, int Cin, int CinP) {
  int t = blockIdx.x * blockDim.x + threadIdx.x;
  int total = Cout * CinP;
  if (t >= total) return;
  int i = t & 1;
  int lane = (t >> 1) & 31;
  int blk = t >> 6;
  int kt = blk % (CinP >> 2);
  int mtile = blk / (CinP >> 2);
  int row = mtile * 16 + (lane & 15);
  int k = kt * 4 + (lane >> 4) * 2 + i;
  Wp[t] = (k < Cin) ? W[(size_t)row * Cin + k] : 0.0f;
}

// Interleave xs1/xs2/xs3 -> (24, 4, 512, 64), channel 3 zero-padded.
__global__ void build_xs_k(const float* __restrict__ xs1, const float* __restrict__ xs2,
                           const float* __restrict__ xs3, float* __restrict__ out) {
  size_t t = (size_t)blockIdx.x * blockDim.x + threadIdx.x;
  const size_t total = (size_t)24 * 4 * 32768;
  if (t >= total) return;
  int p = (int)(t % 32768);
  int c = (int)((t / 32768) % 4);
  int xb = (int)(t / (4 * 32768));
  int b = xb / 3, s = xb % 3;
  float v = 0.0f;
  if (c < 3) {
    const float* src = (s == 0) ? xs1 : ((s == 1) ? xs2 : xs3);
    v = src[((size_t)b * 3 + c) * 32768 + p];
  }
  out[t] = v;
}

// Write xyz rows 0..2 of the points buffer from y*, zero pad row 131.
__global__ void init_points_k(float* __restrict__ Pt, const float* __restrict__ y1,
                              const float* __restrict__ y2, const float* __restrict__ y3,
                              int Bx, int isS) {
  int t = blockIdx.x * blockDim.x + threadIdx.x;
  int total = Bx * 4 * 512;
  if (t >= total) return;
  int n = t % 512;
  int c = (t / 512) % 4;
  int xb = t / (4 * 512);
  float v = 0.0f;
  int row = 131;
  if (c < 3) {
    row = c;
    if (isS) {
      int b = xb / 3, s = xb % 3;
      const float* src = (s == 0) ? y1 : ((s == 1) ? y2 : y3);
      v = src[((size_t)b * 3 + c) * 512 + n];
    } else {
      v = y1[((size_t)xb * 3 + c) * 512 + n];
    }
  }
  Pt[((size_t)xb * 132 + row) * 512 + n] = v;
}

// Max over the 128 positions -> transposed feature matrix xT (4096 x 8).
__global__ void pool_feat_k(const float* __restrict__ Y, float* __restrict__ xT,
                            int Bx, int isS) {
  int t = blockIdx.x * blockDim.x + threadIdx.x;
  if (t >= Bx * 1024) return;
  int c = t % 1024;
  int xb = t / 1024;
  const float* p = Y + (size_t)t * 128;
  float mx = p[0];
  for (int i = 1; i < 128; ++i) mx = fmaxf(mx, p[i]);
  int row, col;
  if (isS) { row = 1024 + (xb % 3) * 1024 + c; col = xb / 3; }
  else { row = c; col = xb; }
  xT[row * 8 + col] = mx;
}

// Final tiny layer: out(8,60) = y1(256,8)^T @ W2^T + b2.
__global__ void fc2_k(const float* __restrict__ W2, const float* __restrict__ b2,
                      const float* __restrict__ y1, float* __restrict__ out) {
  int t = blockIdx.x * blockDim.x + threadIdx.x;
  if (t >= 480) return;
  int b = t / 60, o = t % 60;
  float s = b2[o];
  for (int c = 0; c < 256; ++c) s += W2[o * 256 + c] * y1[c * 8 + b];
  out[b * 60 + o] = s;
}

// ---------------------------------------------------------------------------
// Host orchestration
// ---------------------------------------------------------------------------
extern "C" void kernel_launch(void* const* d_in, const int* in_sizes, int n_in,
                              void* d_out, int out_size, void* d_ws, size_t ws_size,
                              hipStream_t stream) {
  (void)n_in; (void)out_size;
  // Detect flattening order: insertion order puts xt (1,048,576 elems) first;
  // sorted-pytree order puts params.c1.W0 (192 elems) first.
  const bool ins = (in_sizes[0] == 1048576);
  int I_xt, I_xs1, I_xs2, I_xs3, I_yt, I_ys1, I_ys2, I_ys3;
  int I_t1, I_t2, I_t3, I_c1, I_c2, I_c3, I_fc;
  if (ins) {
    I_xt = 0; I_xs1 = 1; I_xs2 = 2; I_xs3 = 3;
    I_yt = 4; I_ys1 = 5; I_ys2 = 6; I_ys3 = 7;
    I_t1 = 8; I_t2 = 14; I_t3 = 20; I_c1 = 26; I_c2 = 32; I_c3 = 38; I_fc = 44;
  } else {
    I_c1 = 0; I_c2 = 6; I_c3 = 12; I_fc = 18; I_t1 = 22; I_t2 = 28; I_t3 = 34;
    I_xs1 = 40; I_xs2 = 41; I_xs3 = 42; I_xt = 43;
    I_ys1 = 44; I_ys2 = 45; I_ys3 = 46; I_yt = 47;
  }
  auto F = [&](int i) { return (const float*)d_in[i]; };

  // ---- workspace layout (floats) ----
  float* ws = (float*)d_ws;
  size_t off = 0;
  auto alloc = [&](size_t n) -> float* {
    float* p = ws + off;
    off += (n + 63) & ~(size_t)63;
    return p;
  };
  float* xs_in = alloc((size_t)24 * 4 * 512 * 64);   // interleaved s-branch input
  float* Pt_t  = alloc((size_t)8 * 132 * 512);       // t-branch points (xyz+feat)
  float* Pt_s  = alloc((size_t)24 * 132 * 512);
  float* buf_t = alloc((size_t)8 * 260 * 128);       // ctr + pooled stage2 (259 rows + pad)
  float* buf_s = alloc((size_t)24 * 260 * 128);
  float* act_a = alloc((size_t)24 * 256 * 128);      // stage3 activations (shared t/s)
  float* act_b = alloc((size_t)24 * 512 * 128);
  float* act_c = alloc((size_t)24 * 1024 * 128);
  float* xT    = alloc((size_t)4096 * 8);            // concat features, transposed
  float* y1fc  = alloc((size_t)256 * 8);

  // packed weights
  float* pw[19];
  struct WD { int idx, cout, cin, cinp; };
  const WD wd[19] = {
      {I_t1 + 0, 64, 4, 4},     {I_t1 + 2, 64, 64, 64},   {I_t1 + 4, 128, 64, 64},
      {I_t2 + 0, 128, 131, 132},{I_t2 + 2, 128, 128, 128},{I_t2 + 4, 256, 128, 128},
      {I_t3 + 0, 256, 259, 260},{I_t3 + 2, 512, 256, 256},{I_t3 + 4, 1024, 512, 512},
      {I_c1 + 0, 64, 3, 4},     {I_c1 + 2, 64, 64, 64},   {I_c1 + 4, 128, 64, 64},
      {I_c2 + 0, 128, 131, 132},{I_c2 + 2, 128, 128, 128},{I_c2 + 4, 256, 128, 128},
      {I_c3 + 0, 256, 259, 260},{I_c3 + 2, 512, 256, 256},{I_c3 + 4, 1024, 512, 512},
      {I_fc + 0, 256, 4096, 4096}};
  for (int i = 0; i < 19; ++i) pw[i] = alloc((size_t)wd[i].cout * wd[i].cinp);
  if (ws_size < off * sizeof(float)) return;  // insufficient scratch; bail safely

  for (int i = 0; i < 19; ++i) {
    int total = wd[i].cout * wd[i].cinp;
    pack_w_k<<<(total + 255) / 256, 256, 0, stream>>>(F(wd[i].idx), pw[i], wd[i].cout,
                                                      wd[i].cin, wd[i].cinp);
  }

  // ---- build inputs ----
  {
    size_t total = (size_t)24 * 4 * 32768;
    build_xs_k<<<(unsigned)((total + 255) / 256), 256, 0, stream>>>(F(I_xs1), F(I_xs2),
                                                                    F(I_xs3), xs_in);
  }
  init_points_k<<<(8 * 4 * 512 + 255) / 256, 256, 0, stream>>>(Pt_t, F(I_yt), F(I_yt),
                                                               F(I_yt), 8, 0);
  init_points_k<<<(24 * 4 * 512 + 255) / 256, 256, 0, stream>>>(Pt_s, F(I_ys1), F(I_ys2),
                                                                F(I_ys3), 24, 1);

  // ---- stage 1: fused 3-layer MLP + max over k ----
  // LDS: ((4+64+64+128)/4) slices * 520 dwords = 135,200 B
  const size_t shm1 = (size_t)((4 + 64 + 64 + 128) >> 2) * 520 * sizeof(float);
  mlp3_pool_k<<<dim3(256, 8), 256, shm1, stream>>>(
      F(I_xt), Pt_t, pw[0], F(I_t1 + 1), 64, pw[1], F(I_t1 + 3), 64, pw[2], F(I_t1 + 5),
      128, 4, 512, 512, 132 * 512, 3);
  mlp3_pool_k<<<dim3(256, 24), 256, shm1, stream>>>(
      xs_in, Pt_s, pw[9], F(I_c1 + 1), 64, pw[10], F(I_c1 + 3), 64, pw[11], F(I_c1 + 5),
      128, 4, 512, 512, 132 * 512, 3);

  // ---- stage 2: fused ball-query group + 3-layer MLP + max over slots ----
  // LDS: ((132+128+128)/4)*264 + 512 + 4 + 64 dwords = 104,752 B
  const size_t shm2 =
      (size_t)(((132 + 128 + 128) >> 2) * 264 + 512 + 4 + 64) * sizeof(float);
  group_mlp3_k<<<dim3(128, 8), 256, shm2, stream>>>(
      Pt_t, buf_t, pw[3], F(I_t2 + 1), 128, pw[4], F(I_t2 + 3), 128, pw[5], F(I_t2 + 5),
      256);
  group_mlp3_k<<<dim3(128, 24), 256, shm2, stream>>>(
      Pt_s, buf_s, pw[12], F(I_c2 + 1), 128, pw[13], F(I_c2 + 3), 128, pw[14],
      F(I_c2 + 5), 256);

  // ---- stage 3: unfused WMMA GEMMs (tiny traffic) + feature max-pool ----
  // t-branch (t3)
  gemm_glb_k<<<dim3(1, 16, 8), 128, 0, stream>>>(pw[6], F(I_t3 + 1), buf_t, act_a, 256,
                                                 260, 128, 1);
  gemm_glb_k<<<dim3(1, 32, 8), 128, 0, stream>>>(pw[7], F(I_t3 + 3), act_a, act_b, 512,
                                                 256, 128, 1);
  gemm_glb_k<<<dim3(1, 64, 8), 128, 0, stream>>>(pw[8], F(I_t3 + 5), act_b, act_c, 1024,
                                                 512, 128, 1);
  pool_feat_k<<<32, 256, 0, stream>>>(act_c, xT, 8, 0);
  // s-branch (c3)
  gemm_glb_k<<<dim3(1, 16, 24), 128, 0, stream>>>(pw[15], F(I_c3 + 1), buf_s, act_a, 256,
                                                  260, 128, 1);
  gemm_glb_k<<<dim3(1, 32, 24), 128, 0, stream>>>(pw[16], F(I_c3 + 3), act_a, act_b, 512,
                                                  256, 128, 1);
  gemm_glb_k<<<dim3(1, 64, 24), 128, 0, stream>>>(pw[17], F(I_c3 + 5), act_b, act_c,
                                                  1024, 512, 128, 1);
  pool_feat_k<<<96, 256, 0, stream>>>(act_c, xT, 24, 1);

  // ---- fc head ----
  gemm_glb_k<<<dim3(1, 16, 1), 128, 0, stream>>>(pw[18], F(I_fc + 1), xT, y1fc, 256,
                                                 4096, 8, 1);
  fc2_k<<<2, 256, 0, stream>>>(F(I_fc + 2), F(I_fc + 3), y1fc, (float*)d_out);
}